// Model_78271484002488
// MI455X (gfx1250) — compile-verified
//
#include <hip/hip_runtime.h>

typedef __attribute__((ext_vector_type(2))) float v2f;
typedef __attribute__((ext_vector_type(8))) float v8f;
typedef unsigned int u32;
typedef __attribute__((ext_vector_type(4))) u32 v4u;
typedef __attribute__((ext_vector_type(8))) int v8i;
typedef __attribute__((ext_vector_type(4))) int v4i;

#define B_   8
#define L_   512
#define D_   32
#define LAT_ 128
#define HM_  4
#define DK_  32
#define NP_  16
#define NR_  32
#define FF_  128
#define PRED_ 96

#if __has_builtin(__builtin_amdgcn_tensor_load_to_lds) && \
    __has_builtin(__builtin_amdgcn_s_wait_tensorcnt)
#define HAVE_TDM 1
#else
#define HAVE_TDM 0
#endif

__device__ __forceinline__ v8f wmma4(v2f a, v2f b, v8f c) {
  // D = A(16x4 f32) * B(4x16 f32) + C(16x16 f32)
  return __builtin_amdgcn_wmma_f32_16x16x4_f32(false, a, false, b, (short)0, c,
                                               false, false);
}
__device__ __forceinline__ v8f vzero8() {
  v8f z = {0.f, 0.f, 0.f, 0.f, 0.f, 0.f, 0.f, 0.f};
  return z;
}

#if HAVE_TDM
// 1D contiguous copy global -> LDS via Tensor Data Mover.
// nelem f32 elements; optional LDS padding: pad (amt_code+1) DWORDs every
// 2^(iv_code+1) DWORDs (hardware D# pad feature) for bank-conflict-free rows.
__device__ __forceinline__ void tdm_copy_1d(const float* gsrc, u32 lds_addr,
                                            int nelem, int pad_en, int iv_code,
                                            int amt_code) {
  unsigned long long ga = (unsigned long long)(size_t)gsrc;
  v4u g0;
  g0.x = 1u;                                        // count=1, user mode
  g0.y = lds_addr;                                  // lds_addr (bytes)
  g0.z = (u32)(ga & 0xffffffffu);                   // global_addr[31:0]
  g0.w = (u32)((ga >> 32) & 0x01ffffffu) | (2u << 30);  // addr[56:32], type=2
  v8i g1;
  g1[0] = 0x00020000 | (pad_en << 20) | (iv_code << 22) | (amt_code << 25);
  g1[1] = (nelem & 0xffff) << 16;                   // tensor_dim0 lo
  g1[2] = ((nelem >> 16) & 0xffff) | (1 << 16);     // tensor_dim0 hi | dim1=1
  g1[3] = (nelem & 0xffff) << 16;                   // tile_dim0
  g1[4] = 1;                                        // tile_dim1 = 1
  g1[5] = nelem;                                    // tensor_dim0_stride lo
  g1[6] = 0;
  g1[7] = 0;
  v4i gz4 = {0, 0, 0, 0};
  v8i gz8 = {0, 0, 0, 0, 0, 0, 0, 0};
  __builtin_amdgcn_tensor_load_to_lds(g0, g1, gz4, gz4, gz8, 0);
}
#endif

// ---------------------------------------------------------------- obs_valid
__global__ void ov_kernel(const float* __restrict__ xmask, float* __restrict__ ov) {
  int idx = blockIdx.x * blockDim.x + threadIdx.x;
  if (idx < B_ * L_) {
    float s = 0.f;
    for (int d = 0; d < D_; ++d) s += xmask[(size_t)idx * D_ + d];
    ov[idx] = (s > 0.f) ? 1.f : 0.f;
  }
}

// --------------------------------------- TE(t) @ W  (rows x 128 @ 128x128)
// mode 0: t = tsrc[row] ; mode 1: t = row * 512/511 (refs linspace)
// tN == 0: row-major out[row*128+c]; tN > 0: transposed out[c*tN + row]
__global__ __launch_bounds__(256) void te_gemm_kernel(
    const float* __restrict__ tsrc, int mode, const float* __restrict__ W,
    float* __restrict__ outp, int tN) {
  __shared__ float TEs[16][LAT_];
  int tid = threadIdx.x;
  int lane = tid & 31, w = tid >> 5;
  int row0 = blockIdx.x * 16;
  for (int i = tid; i < 16 * LAT_; i += 256) {
    int r = i >> 7, c = i & 127;
    float t = (mode == 0) ? tsrc[row0 + r]
                          : (float)(row0 + r) * (512.0f / 511.0f);
    float div = __expf(-(float)(c & ~1) * (9.210340371976184f / 128.0f));
    float ang = t * div;
    TEs[r][c] = (c & 1) ? __cosf(ang) : __sinf(ang);
  }
  __syncthreads();
  int lane15 = lane & 15, half = lane >> 4;
  v8f acc = vzero8();
  for (int kb = 0; kb < 32; ++kb) {
    v2f a, bb;
    int k0 = kb * 4 + half * 2;
    a.x = TEs[lane15][k0];
    a.y = TEs[lane15][k0 + 1];
    bb.x = W[(size_t)k0 * LAT_ + w * 16 + lane15];
    bb.y = W[(size_t)(k0 + 1) * LAT_ + w * 16 + lane15];
    acc = wmma4(a, bb, acc);
  }
  for (int v = 0; v < 8; ++v) {
    int m = v + half * 8;
    int c = w * 16 + lane15;
    if (tN)
      outp[(size_t)c * tN + row0 + m] = acc[v];
    else
      outp[(size_t)(row0 + m) * LAT_ + c] = acc[v];
  }
}

// ------------------------------------------------- fused patch attention
// One block per (b, patch). 8 waves. Fuses scores->exp->masked softmax as
// two GEMMs (num = E@(mask*v), den = E@mask) + W_out projection + pos-embed.
// KmatT layout: [h*32+e][b*512+l] (transposed for coalesced B loads).
#define QIDX(r, c) ((r) * 132 + (c))
#define EIDX(r, c) ((r) * 68 + (c))
__global__ __launch_bounds__(256) void attn_kernel(
    const float* __restrict__ x, const float* __restrict__ xmask,
    const float* __restrict__ xmark, const float* __restrict__ ov,
    const float* __restrict__ KmatT, const float* __restrict__ Qall,
    const float* __restrict__ W_out, const float* __restrict__ b_out,
    float* __restrict__ z) {
  __shared__ float Qs[32 * 132];   // (r, h*32+e), rows padded to 132
  __shared__ float Es[128 * 68];   // exp(scores) rows=(h*32+r), padded 68
  __shared__ float Vms[64 * 32];   // x chunk -> masked values
  __shared__ float Mms[64 * 32];   // mask chunk -> mask
  int tid = threadIdx.x, lane = tid & 31, w = tid >> 5;
  int b = blockIdx.x >> 4, patch = blockIdx.x & 15;
  int lane15 = lane & 15, half = lane >> 4;

#if HAVE_TDM
  if (w == 0) {
    tdm_copy_1d(Qall + (size_t)patch * 32 * LAT_, (u32)(size_t)&Qs[0],
                32 * 128, 1, 6, 3);  // pad 4 DWORDs every 128 -> 132/row
    __builtin_amdgcn_s_wait_tensorcnt(0);
  }
#else
  for (int i = tid; i < 32 * 128; i += 256)
    Qs[QIDX(i >> 7, i & 127)] = Qall[(size_t)patch * 32 * LAT_ + i];
#endif
  v8f accN[2], accD[2];
  accN[0] = vzero8(); accN[1] = vzero8();
  accD[0] = vzero8(); accD[1] = vzero8();
  __syncthreads();

  for (int ch = 0; ch < 8; ++ch) {
    int l0 = ch * 64;
    size_t base = (size_t)(b * L_ + l0) * D_;
    // stage raw x / x_mask chunk into LDS (DMA overlaps the score WMMAs)
#if HAVE_TDM
    if (w == 0) {
      tdm_copy_1d(x + base, (u32)(size_t)&Vms[0], 64 * 32, 0, 0, 0);
      tdm_copy_1d(xmask + base, (u32)(size_t)&Mms[0], 64 * 32, 0, 0, 0);
    }
#else
    for (int i = tid; i < 64 * 32; i += 256) {
      Vms[i] = x[base + i];
      Mms[i] = xmask[base + i];
    }
#endif
    // scores: per head Q_h(32x32) @ K_h^T(32x64-chunk), then exp -> Es
    for (int q = 0; q < 4; ++q) {
      int tile = w * 4 + q;
      int h = tile >> 3, mt = (tile >> 2) & 1, nt = tile & 3;
      int lglob = l0 + nt * 16 + lane15;
      v8f acc = vzero8();
      for (int kb = 0; kb < 8; ++kb) {
        int k0 = kb * 4 + half * 2;
        v2f a, bb;
        a.x = Qs[QIDX(mt * 16 + lane15, h * 32 + k0)];
        a.y = Qs[QIDX(mt * 16 + lane15, h * 32 + k0 + 1)];
        bb.x = KmatT[(size_t)(h * 32 + k0) * 4096 + b * L_ + lglob];
        bb.y = KmatT[(size_t)(h * 32 + k0 + 1) * 4096 + b * L_ + lglob];
        acc = wmma4(a, bb, acc);
      }
      for (int v = 0; v < 8; ++v) {
        int row = h * 32 + mt * 16 + v + half * 8;
        Es[EIDX(row, nt * 16 + lane15)] =
            __expf(acc[v] * 0.17677669529663687f);
      }
    }
#if HAVE_TDM
    if (w == 0) __builtin_amdgcn_s_wait_tensorcnt(0);
#endif
    __syncthreads();
    // in-LDS fixup: apply mask * in_patch * obs_valid
    for (int i = tid; i < 64 * 32; i += 256) {
      int ll = i >> 5;
      int l = l0 + ll;
      float tv = xmark[b * L_ + l];
      float lo = 32.0f * (float)patch, hi = lo + 32.0f;
      float inp = (tv >= lo && tv <= hi && ov[b * L_ + l] > 0.5f) ? 1.f : 0.f;
      float m = Mms[i] * inp;
      Mms[i] = m;
      Vms[i] *= m;
    }
    __syncthreads();
    // num/den accumulation: E(128 x 64) @ Vms/Mms(64 x 32)
    for (int kb = 0; kb < 16; ++kb) {
      int k0 = kb * 4 + half * 2;
      v2f a;
      a.x = Es[EIDX(w * 16 + lane15, k0)];
      a.y = Es[EIDX(w * 16 + lane15, k0 + 1)];
      for (int nt = 0; nt < 2; ++nt) {
        v2f bn, bm;
        int col = nt * 16 + lane15;
        bn.x = Vms[k0 * 32 + col];       bn.y = Vms[(k0 + 1) * 32 + col];
        bm.x = Mms[k0 * 32 + col];       bm.y = Mms[(k0 + 1) * 32 + col];
        accN[nt] = wmma4(a, bn, accN[nt]);
        accD[nt] = wmma4(a, bm, accD[nt]);
      }
    }
    __syncthreads();
  }

  // attn = num/den, re-layout to (r, h*32+d) reusing Qs storage
  for (int nt = 0; nt < 2; ++nt)
    for (int v = 0; v < 8; ++v) {
      int row = w * 16 + v + half * 8;
      int h = row >> 5, r = row & 31;
      int d = nt * 16 + lane15;
      Qs[QIDX(r, h * 32 + d)] = accN[nt][v] / fmaxf(accD[nt][v], 1e-20f);
    }
  __syncthreads();

  // rp = attn(32x128) @ W_out(128x32) + b_out, + pos-embed, scatter to z
  if (w < 4) {
    int mt = w >> 1, nt = w & 1;
    v8f acc = vzero8();
    for (int kb = 0; kb < 32; ++kb) {
      int k0 = kb * 4 + half * 2;
      v2f a, bb;
      a.x = Qs[QIDX(mt * 16 + lane15, k0)];
      a.y = Qs[QIDX(mt * 16 + lane15, k0 + 1)];
      bb.x = W_out[(size_t)k0 * D_ + nt * 16 + lane15];
      bb.y = W_out[(size_t)(k0 + 1) * D_ + nt * 16 + lane15];
      acc = wmma4(a, bb, acc);
    }
    for (int v = 0; v < 8; ++v) {
      int r = mt * 16 + v + half * 8;
      int d = nt * 16 + lane15;
      float div = __expf(-(float)(r & ~1) * (9.210340371976184f / 32.0f));
      float ang = (float)patch * div;
      float pe = (r & 1) ? __cosf(ang) : __sinf(ang);
      z[((size_t)(b * D_ + d) * NP_ + patch) * NR_ + r] =
          acc[v] + b_out[d] + pe;
    }
  }
}

// -------------------------------------- transformer layer (seq 16, dm 32)
__global__ __launch_bounds__(128) void layer_kernel(
    float* __restrict__ z, const float* __restrict__ Wq,
    const float* __restrict__ Wk, const float* __restrict__ Wv,
    const float* __restrict__ Wo, const float* __restrict__ bq,
    const float* __restrict__ bk, const float* __restrict__ bv,
    const float* __restrict__ bo, const float* __restrict__ ln1g,
    const float* __restrict__ ln1b, const float* __restrict__ W1,
    const float* __restrict__ fb1, const float* __restrict__ W2,
    const float* __restrict__ fb2, const float* __restrict__ ln2g,
    const float* __restrict__ ln2b) {
  int n = blockIdx.x, tid = threadIdx.x;
  __shared__ float zs[16][32], qs[16][32], ks[16][32], vs[16][32];
  __shared__ float as_[8][16][16];
  __shared__ float tmp[16][32];
  __shared__ float y1[16][128];

  for (int i = tid; i < 512; i += 128) zs[i >> 5][i & 31] = z[(size_t)n * 512 + i];
  __syncthreads();
  for (int i = tid; i < 512; i += 128) {
    int s = i >> 5, c = i & 31;
    float aq = bq[c], ak = bk[c], av = bv[c];
    for (int k = 0; k < 32; ++k) {
      float zk = zs[s][k];
      aq += zk * Wq[k * 32 + c];
      ak += zk * Wk[k * 32 + c];
      av += zk * Wv[k * 32 + c];
    }
    qs[s][c] = aq; ks[s][c] = ak; vs[s][c] = av;
  }
  __syncthreads();
  {
    int h = tid >> 4, qi = tid & 15;
    float sc[16], mx = -1e30f;
    for (int j = 0; j < 16; ++j) {
      float dot = 0.f;
      for (int e = 0; e < 4; ++e) dot += qs[qi][h * 4 + e] * ks[j][h * 4 + e];
      sc[j] = dot * 0.5f;
      mx = fmaxf(mx, sc[j]);
    }
    float sum = 0.f;
    for (int j = 0; j < 16; ++j) { sc[j] = __expf(sc[j] - mx); sum += sc[j]; }
    float inv = 1.f / sum;
    for (int j = 0; j < 16; ++j) as_[h][qi][j] = sc[j] * inv;
  }
  __syncthreads();
  for (int i = tid; i < 512; i += 128) {
    int s = i >> 5, c = i & 31, h = c >> 2;
    float acc = 0.f;
    for (int j = 0; j < 16; ++j) acc += as_[h][s][j] * vs[j][c];
    tmp[s][c] = acc;
  }
  __syncthreads();
  for (int i = tid; i < 512; i += 128) {
    int s = i >> 5, c = i & 31;
    float acc = bo[c];
    for (int k = 0; k < 32; ++k) acc += tmp[s][k] * Wo[k * 32 + c];
    qs[s][c] = zs[s][c] + acc;  // pre-LN1 buffer
  }
  __syncthreads();
  if (tid < 16) {
    int s = tid;
    float mu = 0.f;
    for (int c = 0; c < 32; ++c) mu += qs[s][c];
    mu *= (1.f / 32.f);
    float var = 0.f;
    for (int c = 0; c < 32; ++c) { float d0 = qs[s][c] - mu; var += d0 * d0; }
    float rs = rsqrtf(var * (1.f / 32.f) + 1e-5f);
    for (int c = 0; c < 32; ++c)
      zs[s][c] = (qs[s][c] - mu) * rs * ln1g[c] + ln1b[c];
  }
  __syncthreads();
  for (int i = tid; i < 16 * 128; i += 128) {
    int s = i >> 7, f = i & 127;
    float acc = fb1[f];
    for (int k = 0; k < 32; ++k) acc += zs[s][k] * W1[k * 128 + f];
    float t3 = acc + 0.044715f * acc * acc * acc;
    y1[s][f] = 0.5f * acc * (1.f + tanhf(0.7978845608028654f * t3));
  }
  __syncthreads();
  for (int i = tid; i < 512; i += 128) {
    int s = i >> 5, c = i & 31;
    float acc = fb2[c];
    for (int f = 0; f < 128; ++f) acc += y1[s][f] * W2[f * 32 + c];
    tmp[s][c] = zs[s][c] + acc;
  }
  __syncthreads();
  if (tid < 16) {
    int s = tid;
    float mu = 0.f;
    for (int c = 0; c < 32; ++c) mu += tmp[s][c];
    mu *= (1.f / 32.f);
    float var = 0.f;
    for (int c = 0; c < 32; ++c) { float d0 = tmp[s][c] - mu; var += d0 * d0; }
    float rs = rsqrtf(var * (1.f / 32.f) + 1e-5f);
    for (int c = 0; c < 32; ++c)
      z[(size_t)n * 512 + s * 32 + c] = (tmp[s][c] - mu) * rs * ln2g[c] + ln2b[c];
  }
}

// ------------------------- final LN + transpose-reshape into emb(256x512)
__global__ void lnf_kernel(const float* __restrict__ z, const float* __restrict__ g,
                           const float* __restrict__ bb, float* __restrict__ emb) {
  int n = blockIdx.x, tid = threadIdx.x;
  if (tid < 16) {
    int p = tid;
    float buf[32], mu = 0.f;
    for (int r = 0; r < 32; ++r) {
      buf[r] = z[(size_t)n * 512 + p * 32 + r];
      mu += buf[r];
    }
    mu *= (1.f / 32.f);
    float var = 0.f;
    for (int r = 0; r < 32; ++r) { float d0 = buf[r] - mu; var += d0 * d0; }
    float rs = rsqrtf(var * (1.f / 32.f) + 1e-5f);
    for (int r = 0; r < 32; ++r)
      emb[(size_t)n * 512 + r * 16 + p] = (buf[r] - mu) * rs * g[r] + bb[r];
  }
}

// ---------------------------- out = emb(256x512) @ W_lin(512x96) + b_lin
// One block per 16-row panel; panel staged once (TDM), 6 waves = 6 col tiles.
#define AIDX(r, k) ((r) * 520 + (k) + (((k) >> 8) * 4))
__global__ __launch_bounds__(192) void out_gemm_kernel(
    const float* __restrict__ emb, const float* __restrict__ W_lin,
    const float* __restrict__ b_lin, float* __restrict__ out) {
  __shared__ float As[16 * 520];  // 16 rows x 512, pad 4 DWORDs every 256
  int tid = threadIdx.x, lane = tid & 31, w = tid >> 5;  // w = col tile
  int mt = blockIdx.x;
  int lane15 = lane & 15, half = lane >> 4;
#if HAVE_TDM
  if (w == 0) {
    tdm_copy_1d(emb + (size_t)mt * 16 * 512, (u32)(size_t)&As[0], 16 * 512, 1,
                7, 3);  // pad 4 DWORDs every 256 -> 520/row
    __builtin_amdgcn_s_wait_tensorcnt(0);
  }
#else
  for (int i = tid; i < 16 * 512; i += 192)
    As[AIDX(i >> 9, i & 511)] = emb[(size_t)mt * 16 * 512 + i];
#endif
  __syncthreads();
  v8f acc = vzero8();
  for (int kb = 0; kb < 128; ++kb) {
    int k0 = kb * 4 + half * 2;
    v2f a, bb;
    a.x = As[AIDX(lane15, k0)];
    a.y = As[AIDX(lane15, k0 + 1)];
    bb.x = W_lin[(size_t)k0 * PRED_ + w * 16 + lane15];
    bb.y = W_lin[(size_t)(k0 + 1) * PRED_ + w * 16 + lane15];
    acc = wmma4(a, bb, acc);
  }
  for (int v = 0; v < 8; ++v) {
    int n = mt * 16 + v + half * 8;
    int b = n >> 5, d = n & 31;
    int j = w * 16 + lane15;
    out[((size_t)b * PRED_ + j) * D_ + d] = acc[v] + b_lin[j];
  }
}

extern "C" void kernel_launch(void* const* d_in, const int* in_sizes, int n_in,
                              void* d_out, int out_size, void* d_ws, size_t ws_size,
                              hipStream_t stream) {
  const float* x      = (const float*)d_in[0];
  const float* xmark  = (const float*)d_in[1];
  const float* xmask  = (const float*)d_in[2];
  const float* Wq     = (const float*)d_in[3];
  const float* Wk     = (const float*)d_in[4];
  const float* W_out  = (const float*)d_in[5];
  const float* b_out  = (const float*)d_in[6];
  const float* f_Wq   = (const float*)d_in[7];
  const float* f_Wk   = (const float*)d_in[8];
  const float* f_Wv   = (const float*)d_in[9];
  const float* f_Wo   = (const float*)d_in[10];
  const float* f_bq   = (const float*)d_in[11];
  const float* f_bk   = (const float*)d_in[12];
  const float* f_bv   = (const float*)d_in[13];
  const float* f_bo   = (const float*)d_in[14];
  const float* f_ln1g = (const float*)d_in[15];
  const float* f_ln1b = (const float*)d_in[16];
  const float* f_W1   = (const float*)d_in[17];
  const float* f_b1   = (const float*)d_in[18];
  const float* f_W2   = (const float*)d_in[19];
  const float* f_b2   = (const float*)d_in[20];
  const float* f_ln2g = (const float*)d_in[21];
  const float* f_ln2b = (const float*)d_in[22];
  const float* ln_f_g = (const float*)d_in[23];
  const float* ln_f_b = (const float*)d_in[24];
  const float* W_lin  = (const float*)d_in[25];
  const float* b_lin  = (const float*)d_in[26];
  float* out = (float*)d_out;

  float* ws    = (float*)d_ws;
  float* KmatT = ws;                                  // 128 * 4096 (transposed)
  float* Qall  = KmatT + (size_t)128 * 4096;          // 512*128
  float* ov    = Qall + (size_t)512 * 128;            // 4096
  float* z     = ov + 4096;                           // 256*16*32
  float* emb   = z + (size_t)256 * 16 * 32;           // 256*512

  ov_kernel<<<(B_ * L_ + 127) / 128, 128, 0, stream>>>(xmask, ov);
  te_gemm_kernel<<<(B_ * L_) / 16, 256, 0, stream>>>(xmark, 0, Wk, KmatT, 4096);
  te_gemm_kernel<<<(NP_ * NR_) / 16, 256, 0, stream>>>(nullptr, 1, Wq, Qall, 0);
  attn_kernel<<<B_ * NP_, 256, 0, stream>>>(x, xmask, xmark, ov, KmatT, Qall,
                                            W_out, b_out, z);
  for (int l = 0; l < 3; ++l) {
    layer_kernel<<<B_ * D_, 128, 0, stream>>>(
        z, f_Wq + l * 1024, f_Wk + l * 1024, f_Wv + l * 1024, f_Wo + l * 1024,
        f_bq + l * 32, f_bk + l * 32, f_bv + l * 32, f_bo + l * 32,
        f_ln1g + l * 32, f_ln1b + l * 32, f_W1 + l * 32 * 128, f_b1 + l * 128,
        f_W2 + l * 128 * 32, f_b2 + l * 32, f_ln2g + l * 32, f_ln2b + l * 32);
  }
  lnf_kernel<<<B_ * D_, 32, 0, stream>>>(z, ln_f_g, ln_f_b, emb);
  out_gemm_kernel<<<16, 192, 0, stream>>>(emb, W_lin, b_lin, out);
}